// ChamferLoss_86294482911473
// MI455X (gfx1250) — compile-verified
//
#include <hip/hip_runtime.h>
#include <hip/hip_bf16.h>

// Chamfer loss via V_WMMA_F32_16X16X4_F32 on gfx1250 (wave32).
//
// sq(m,n) = ||a_m||^2 + ||c_n||^2 - 2 a_m . c_n
// Pack A rows as (-2x,-2y,-2z, 1) and B cols as (x, y, z, ||c||^2): one K=4 f32
// WMMA yields a 16x16 tile of t(m,n) = ||c_n||^2 - 2 a.c; the row norm
// ||a_m||^2 is a per-row constant added after the column min-reduction.

typedef float v2f __attribute__((ext_vector_type(2)));
typedef float v8f __attribute__((ext_vector_type(8)));

#define BATCH   8
#define NPTS    4096
#define TILES_N (NPTS / 16)   // 256 column tiles per batch
#define WAVES   8
#define BLOCK   (WAVES * 32)  // 256 threads = 8 wave32
#define NITER   (TILES_N / WAVES)  // 32 tiles per wave

// Raw v_min_num_f32: avoids the fminf NaN-canonicalization (v_max x,x,x) pair
// the compiler otherwise inserts around every min in the hot loop.
__device__ __forceinline__ float minraw(float a, float b) {
    float r;
    asm("v_min_num_f32 %0, %1, %2" : "=v"(r) : "v"(a), "v"(b));
    return r;
}

// ---------------------------------------------------------------------------
// Pack y1/y2 ([B*P, 3] f32, row major) into WMMA-friendly float4 arrays:
//   PAi[t] = (-2x, -2y, -2z, 1)          (A-side fragment source)
//   PBi[t] = ( x,   y,   z, x^2+y^2+z^2) (B-side fragment source + row norms)
// ---------------------------------------------------------------------------
__global__ void chamfer_prep(const float* __restrict__ y1,
                             const float* __restrict__ y2,
                             float4* __restrict__ PA1, float4* __restrict__ PB1,
                             float4* __restrict__ PA2, float4* __restrict__ PB2,
                             int n) {
    int t = blockIdx.x * blockDim.x + threadIdx.x;
    if (t >= n) return;
    float x = y1[3 * t + 0], y = y1[3 * t + 1], z = y1[3 * t + 2];
    PA1[t] = make_float4(-2.f * x, -2.f * y, -2.f * z, 1.f);
    PB1[t] = make_float4(x, y, z, x * x + y * y + z * z);
    x = y2[3 * t + 0]; y = y2[3 * t + 1]; z = y2[3 * t + 2];
    PA2[t] = make_float4(-2.f * x, -2.f * y, -2.f * z, 1.f);
    PB2[t] = make_float4(x, y, z, x * x + y * y + z * z);
}

__global__ void chamfer_zero(float* __restrict__ out) {
    if (threadIdx.x == 0 && blockIdx.x == 0) *out = 0.f;
}

// ---------------------------------------------------------------------------
// One direction of the chamfer sum. Block = (batch b, 16-row strip of PA).
// Each of 8 waves scans 32 of the 256 column tiles (unrolled x4: four b64
// loads with immediate offsets feed four WMMAs per pointer bump), keeping an
// elementwise running min. Scalar (readfirstlane) loop counter keeps EXEC
// all-ones and loop control on the SALU.
// ---------------------------------------------------------------------------
__global__ void __launch_bounds__(BLOCK)
chamfer_dir(const float4* __restrict__ PA, const float4* __restrict__ PB,
            const float4* __restrict__ PBrow, float* __restrict__ out,
            float scale) {
    __shared__ float partial[WAVES][16];

    const int lane  = threadIdx.x & 31;
    const int laneN = lane & 15;
    const int hi    = lane >> 4;              // 0: K=0,1 half; 1: K=2,3 half
    const int wave  = __builtin_amdgcn_readfirstlane(threadIdx.x >> 5);
    const int mtile = blockIdx.x & (TILES_N - 1);
    const int b     = blockIdx.x / TILES_N;

    const int rowbase = b * NPTS + mtile * 16;

    // A fragment (fixed for the whole block):
    //   lane<16 : (-2x, -2y) of point rowbase+laneN
    //   lane>=16: (-2z,  1 ) of point rowbase+laneN
    const float* pa = (const float*)(PA + rowbase + laneN) + 2 * hi;
    const v2f afrag = *(const v2f*)pa;

    v8f acc;
#pragma unroll
    for (int i = 0; i < 8; ++i) acc[i] = 3.4e38f;

    // Per-lane B-fragment pointer for this wave's first tile (j = wave).
    // Advancing j by WAVES moves 8*16 points = 512 floats = 2048 bytes.
    const float* pb = (const float*)(PB + (size_t)b * NPTS)
                      + ((wave * 16 + laneN) * 4 + 2 * hi);

#pragma unroll 1
    for (int it = 0; it < NITER; it += 4) {
        // Four independent load->WMMA->min chains; offsets fold into IOFFSET.
        const v2f b0 = *(const v2f*)(pb + 0 * 512);
        const v2f b1 = *(const v2f*)(pb + 1 * 512);
        const v2f b2 = *(const v2f*)(pb + 2 * 512);
        const v2f b3 = *(const v2f*)(pb + 3 * 512);
        pb += 4 * 512;

        v8f c0 = {0.f, 0.f, 0.f, 0.f, 0.f, 0.f, 0.f, 0.f};
        v8f c1 = c0, c2 = c0, c3 = c0;
        c0 = __builtin_amdgcn_wmma_f32_16x16x4_f32(false, afrag, false, b0,
                                                   (short)0, c0, false, false);
        c1 = __builtin_amdgcn_wmma_f32_16x16x4_f32(false, afrag, false, b1,
                                                   (short)0, c1, false, false);
        c2 = __builtin_amdgcn_wmma_f32_16x16x4_f32(false, afrag, false, b2,
                                                   (short)0, c2, false, false);
        c3 = __builtin_amdgcn_wmma_f32_16x16x4_f32(false, afrag, false, b3,
                                                   (short)0, c3, false, false);
#pragma unroll
        for (int i = 0; i < 8; ++i) {
            acc[i] = minraw(acc[i], minraw(minraw(c0[i], c1[i]),
                                           minraw(c2[i], c3[i])));
        }
    }

    // Min-reduce across the 16 lanes of each half (rows i / i+8 per C/D
    // layout; xor masks 1,2,4,8 never cross the 16-lane boundary).
#pragma unroll
    for (int m = 1; m < 16; m <<= 1) {
#pragma unroll
        for (int i = 0; i < 8; ++i)
            acc[i] = minraw(acc[i], __shfl_xor(acc[i], m, 32));
    }

    if (laneN == 0) {
#pragma unroll
        for (int i = 0; i < 8; ++i) partial[wave][hi * 8 + i] = acc[i];
    }
    __syncthreads();

    if (threadIdx.x < 16) {
        const int r = threadIdx.x;
        float t = partial[0][r];
#pragma unroll
        for (int w = 1; w < WAVES; ++w) t = minraw(t, partial[w][r]);
        const int row = rowbase + r;
        const float aa = ((const float*)(PBrow + row))[3];  // ||a_m||^2
        const float d = sqrtf(fmaxf(aa + t, 0.f));
        atomicAdd(out, d * scale);
    }
}

extern "C" void kernel_launch(void* const* d_in, const int* in_sizes, int n_in,
                              void* d_out, int out_size, void* d_ws, size_t ws_size,
                              hipStream_t stream) {
    const float* y1 = (const float*)d_in[0];
    const float* y2 = (const float*)d_in[1];
    // d_in[2]/d_in[3] (batch ids) encode the contiguous equal-size layout; unused.

    const int n = BATCH * NPTS;

    float4* PA1 = (float4*)d_ws;
    float4* PB1 = PA1 + n;
    float4* PA2 = PB1 + n;
    float4* PB2 = PA2 + n;

    float* out = (float*)d_out;

    chamfer_prep<<<(n + 255) / 256, 256, 0, stream>>>(y1, y2, PA1, PB1, PA2, PB2, n);
    chamfer_zero<<<1, 32, 0, stream>>>(out);

    const float scale = 1.0f / (float)n;
    const int grid = BATCH * TILES_N;  // 2048 blocks per direction
    // d1: rows from y1, cols from y2 (row norms come from PB1.w)
    chamfer_dir<<<grid, BLOCK, 0, stream>>>(PA1, PB2, PB1, out, scale);
    // d2: rows from y2, cols from y1
    chamfer_dir<<<grid, BLOCK, 0, stream>>>(PA2, PB1, PB2, out, scale);
}